// DLRM_Net_25099788878056
// MI455X (gfx1250) — compile-verified
//
#include <hip/hip_runtime.h>
#include <hip/hip_bf16.h>
#include <math.h>

#define B_BATCH 4096
#define N_TAB   26
#define V_ROWS  100000
#define D_EMB   64
#define L_POOL  4
#define T_ROWS  27
#define T_STRIDE (T_ROWS * D_EMB)   // 1728 floats per sample
#define R_DIM   416                 // 415 padded to 416 (= 26*16)
#define Z_PAIRS 351

#define KB   32                     // K-chunk staged in LDS
#define LSTR 36                     // LDS row stride (floats): 16B-aligned rows, bank-friendly

typedef __attribute__((ext_vector_type(2))) float v2f;
typedef __attribute__((ext_vector_type(8))) float v8f;

// ---------------------------------------------------------------------------
// CDNA5 async global->LDS copy (16B per lane), tracked by ASYNCcnt.
// VDST = per-lane LDS byte offset (low 32 bits of a generic LDS pointer,
// per the aperture truncation rule), VADDR = per-lane 64-bit global address.
// No destination VGPRs -> no register pressure for prefetch staging.
// ---------------------------------------------------------------------------
__device__ __forceinline__ void async_copy_b128(const float* g, float* l) {
    unsigned loff = (unsigned)(size_t)(void*)l;
    asm volatile("global_load_async_to_lds_b128 %0, %1, off"
                 :: "v"(loff), "v"(g) : "memory");
}
__device__ __forceinline__ void wait_async_zero() {
    asm volatile("s_wait_asynccnt 0x0" ::: "memory");
}

// ---------------------------------------------------------------------------
// Zero-pad rows: src [M x Kin] -> dst [M x Kpad]
// ---------------------------------------------------------------------------
__global__ void pad_rows(const float* __restrict__ src, float* __restrict__ dst,
                         int M, int Kin, int Kpad) {
    int idx = blockIdx.x * blockDim.x + threadIdx.x;
    if (idx >= M * Kpad) return;
    int r = idx / Kpad, c = idx - r * Kpad;
    dst[idx] = (c < Kin) ? src[(size_t)r * Kin + c] : 0.0f;
}

// ---------------------------------------------------------------------------
// Embedding gather + pool: T[b][1+t][d] = sum_{l<4} emb[t][idx[t][b*4+l]][d]
// 64 threads per (t,b) row; 4 rows per 256-thread block. Each pooled row is a
// 256B coalesced gather -> pure HBM-bandwidth bound, as it should be.
// ---------------------------------------------------------------------------
__global__ void embed_pool(const float* __restrict__ emb,
                           const int* __restrict__ lS_i,
                           float* __restrict__ T) {
    int row = blockIdx.x * 4 + (threadIdx.x >> 6);
    int d   = threadIdx.x & 63;
    int t   = row / B_BATCH;
    int b   = row - t * B_BATCH;
    const int* idx = lS_i + (size_t)t * (B_BATCH * L_POOL) + b * L_POOL;
    float s = 0.0f;
#pragma unroll
    for (int l = 0; l < L_POOL; ++l) {
        size_t off = ((size_t)t * V_ROWS + (size_t)idx[l]) * D_EMB + d;
        s += emb[off];
    }
    T[(size_t)b * T_STRIDE + (size_t)(1 + t) * D_EMB + d] = s;
}

// ---------------------------------------------------------------------------
// OUT[b][n] = ReLU( sum_k IN[b][k] * W[n][k] + bias[n] )
//
// Block = 128 threads (4 waves) -> 64(batch) x 64(feature) C tile.
// K is processed in KB=32 chunks, double-buffered through LDS using CDNA5
// async global->LDS copies (GLOBAL_LOAD_ASYNC_TO_LDS_B128, ASYNCcnt):
//   - chunk c+1's asyncs are issued before compute(c) and waited after it,
//     so the HBM/L2 latency hides under 32 WMMAs; no staging VGPRs, no spills
//   - compute reads v2f operands from LDS (ds_load_b64, low latency)
//
// V_WMMA_F32_16X16X4_F32 operand layout (ISA 7.12.2):
//   A (16x4): lanes 0-15 hold K={kb,kb+1}, lanes 16-31 hold K={kb+2,kb+3}
//   B (4x16): same K split; lane&15 selects N column
//   C/D:      VGPR r -> M = r + 8*(lane>=16), N = lane&15
//
// Requires K % 32 == 0, grid = (B/64, M/64).
// ---------------------------------------------------------------------------
__global__ void gemm_relu_wmma(const float* __restrict__ IN, int ldi,
                               const float* __restrict__ W,   // [M x K] row-major
                               const float* __restrict__ bias,
                               float* __restrict__ OUT, int ldo,
                               int K, int M) {
    __shared__ float sA[2][64 * LSTR];
    __shared__ float sW[2][64 * LSTR];

    const int tid  = threadIdx.x;
    const int lane = tid & 31;
    const int wave = tid >> 5;
    const int half = lane >> 4;
    const int l16  = lane & 15;
    const int b0   = blockIdx.x * 64;
    const int n0   = blockIdx.y * 64;

    // Cooperative fetch mapping: 64 rows x 8 float4-cols = 512 xfers / tile,
    // 128 threads -> 4 x 16B async copies each (per tile).
    int rowi[4], coli[4];
#pragma unroll
    for (int i = 0; i < 4; ++i) {
        int idx = tid + 128 * i;
        rowi[i] = idx >> 3;
        coli[i] = (idx & 7) * 4;
    }

    const int nch = K / KB;

    // Prologue: async-stage chunk 0 into buffer 0.
#pragma unroll
    for (int i = 0; i < 4; ++i) {
        async_copy_b128(IN + (size_t)(b0 + rowi[i]) * ldi + coli[i],
                        &sA[0][rowi[i] * LSTR + coli[i]]);
        async_copy_b128(W  + (size_t)(n0 + rowi[i]) * K   + coli[i],
                        &sW[0][rowi[i] * LSTR + coli[i]]);
    }
    wait_async_zero();
    __syncthreads();

    v8f c0 = {}, c1 = {}, c2 = {}, c3 = {};

    for (int c = 0; c < nch; ++c) {
        const int p = c & 1;
        const int q = p ^ 1;

        // Issue next chunk's async copies (overlap with compute below).
        // Safe: buf q was last read in compute(c-1); every wave passed the
        // barrier at the end of iteration c-1 before reaching this point.
        if (c + 1 < nch) {
            const int kb0 = (c + 1) * KB;
#pragma unroll
            for (int i = 0; i < 4; ++i) {
                async_copy_b128(IN + (size_t)(b0 + rowi[i]) * ldi + kb0 + coli[i],
                                &sA[q][rowi[i] * LSTR + coli[i]]);
                async_copy_b128(W  + (size_t)(n0 + rowi[i]) * K   + kb0 + coli[i],
                                &sW[q][rowi[i] * LSTR + coli[i]]);
            }
        }

        // Compute phase: 8 k-steps x 4 WMMAs from LDS. Wave's A rows are
        // b0 + wave*16 .. +15.
        const float* Al = &sA[p][(wave * 16) * LSTR];
        const float* Wl = &sW[p][0];
#pragma unroll
        for (int ks = 0; ks < KB; ks += 4) {
            const int ko = ks + 2 * half;
            v2f a  = *(const v2f*)(Al + l16 * LSTR + ko);
            v2f w0 = *(const v2f*)(Wl + (l16     ) * LSTR + ko);
            v2f w1 = *(const v2f*)(Wl + (l16 + 16) * LSTR + ko);
            v2f w2 = *(const v2f*)(Wl + (l16 + 32) * LSTR + ko);
            v2f w3 = *(const v2f*)(Wl + (l16 + 48) * LSTR + ko);
            c0 = __builtin_amdgcn_wmma_f32_16x16x4_f32(false, a, false, w0, (short)0, c0, false, false);
            c1 = __builtin_amdgcn_wmma_f32_16x16x4_f32(false, a, false, w1, (short)0, c1, false, false);
            c2 = __builtin_amdgcn_wmma_f32_16x16x4_f32(false, a, false, w2, (short)0, c2, false, false);
            c3 = __builtin_amdgcn_wmma_f32_16x16x4_f32(false, a, false, w3, (short)0, c3, false, false);
        }

        // Make chunk c+1 visible to all waves.
        if (c + 1 < nch) {
            wait_async_zero();   // this wave's async writes have landed in LDS
            __syncthreads();     // all waves' writes visible; reads of buf p done
        }
    }

    // Epilogue: bias + ReLU, scatter per C/D layout.
    const int bm0 = b0 + wave * 16;
    const float bv0 = bias[n0      + l16];
    const float bv1 = bias[n0 + 16 + l16];
    const float bv2 = bias[n0 + 32 + l16];
    const float bv3 = bias[n0 + 48 + l16];
#pragma unroll
    for (int r = 0; r < 8; ++r) {
        const int mrow = bm0 + 8 * half + r;
        float* o = OUT + (size_t)mrow * ldo + n0 + l16;
        float v0 = c0[r] + bv0; o[0]  = v0 > 0.0f ? v0 : 0.0f;
        float v1 = c1[r] + bv1; o[16] = v1 > 0.0f ? v1 : 0.0f;
        float v2 = c2[r] + bv2; o[32] = v2 > 0.0f ? v2 : 0.0f;
        float v3 = c3[r] + bv3; o[48] = v3 > 0.0f ? v3 : 0.0f;
    }
}

// ---------------------------------------------------------------------------
// Feature interaction: block per sample.
//   R[b][0:64]  = T[b][0][:]           (bottom-MLP output x)
//   R[b][64+p]  = <T[b][i], T[b][j]>   for tril pairs (i>j), p = 0..350
//   R[b][415]   = 0                    (padding column)
// ---------------------------------------------------------------------------
__global__ void interact_kernel(const float* __restrict__ T, float* __restrict__ R) {
    __shared__ float s[T_ROWS * 68];
    const int b = blockIdx.x;
    const int tid = threadIdx.x;
    const float* tb = T + (size_t)b * T_STRIDE;
    for (int i = tid; i < T_STRIDE; i += 128) {
        int r = i >> 6, c = i & 63;
        s[r * 68 + c] = tb[i];
    }
    __syncthreads();
    float* rb = R + (size_t)b * R_DIM;
    for (int i = tid; i < 64; i += 128) rb[i] = s[i];   // row 0 == x
    for (int p = tid; p < Z_PAIRS; p += 128) {
        int i = (int)floorf((1.0f + sqrtf(1.0f + 8.0f * (float)p)) * 0.5f);
        while (i * (i - 1) / 2 > p) --i;
        while ((i + 1) * i / 2 <= p) ++i;
        int j = p - i * (i - 1) / 2;
        const float* ri = s + i * 68;
        const float* rj = s + j * 68;
        float acc = 0.0f;
#pragma unroll
        for (int d = 0; d < 64; ++d) acc += ri[d] * rj[d];
        rb[64 + p] = acc;
    }
    if (tid == 0) rb[415] = 0.0f;
}

// ---------------------------------------------------------------------------
// Final 256 -> 1 layer + sigmoid. One wave32 per sample, 8 samples per block.
// ---------------------------------------------------------------------------
__global__ void top_final(const float* __restrict__ A, const float* __restrict__ W,
                          const float* __restrict__ bias, float* __restrict__ out) {
    const int lane = threadIdx.x & 31;
    const int b    = blockIdx.x * 8 + (threadIdx.x >> 5);
    const float* a = A + (size_t)b * 256;
    float sum = 0.0f;
#pragma unroll
    for (int k = lane; k < 256; k += 32) sum += a[k] * W[k];
#pragma unroll
    for (int off = 16; off > 0; off >>= 1) sum += __shfl_xor(sum, off, 32);
    if (lane == 0) out[b] = 1.0f / (1.0f + expf(-(sum + bias[0])));
}

// ---------------------------------------------------------------------------
extern "C" void kernel_launch(void* const* d_in, const int* in_sizes, int n_in,
                              void* d_out, int out_size, void* d_ws, size_t ws_size,
                              hipStream_t stream) {
    const float* dense_x = (const float*)d_in[0];
    // d_in[1] = lS_o: offsets are the constant pattern b*4, not needed
    const int*   lS_i    = (const int*)d_in[2];
    const float* emb_w   = (const float*)d_in[3];
    const float* Wb0 = (const float*)d_in[4];
    const float* bb0 = (const float*)d_in[5];
    const float* Wb1 = (const float*)d_in[6];
    const float* bb1 = (const float*)d_in[7];
    const float* Wb2 = (const float*)d_in[8];
    const float* bb2 = (const float*)d_in[9];
    const float* Wt0 = (const float*)d_in[10];
    const float* bt0 = (const float*)d_in[11];
    const float* Wt1 = (const float*)d_in[12];
    const float* bt1 = (const float*)d_in[13];
    const float* Wt2 = (const float*)d_in[14];
    const float* bt2 = (const float*)d_in[15];
    float* out = (float*)d_out;

    // Workspace carve-up (floats); every region fully written before read.
    float* ws     = (float*)d_ws;
    float* Xpad   = ws;                                   // 4096*32 (13 -> 32)
    float* W0pad  = Xpad   + (size_t)B_BATCH * 32;        // 512*32
    float* Wt0pad = W0pad  + (size_t)512 * 32;            // 512*416
    float* A1     = Wt0pad + (size_t)512 * R_DIM;         // 4096*512
    float* A2     = A1     + (size_t)B_BATCH * 512;       // 4096*256
    float* Tb     = A2     + (size_t)B_BATCH * 256;       // 4096*1728
    float* R      = Tb     + (size_t)B_BATCH * T_STRIDE;  // 4096*416

    // Pad inputs / weights so every GEMM K is a multiple of KB=32.
    pad_rows<<<(B_BATCH * 32 + 255) / 256, 256, 0, stream>>>(dense_x, Xpad, B_BATCH, 13, 32);
    pad_rows<<<(512 * 32 + 255) / 256, 256, 0, stream>>>(Wb0, W0pad, 512, 13, 32);
    pad_rows<<<(512 * R_DIM + 255) / 256, 256, 0, stream>>>(Wt0, Wt0pad, 512, 415, R_DIM);

    // Embedding gather+pool into interaction buffer rows 1..26.
    embed_pool<<<N_TAB * B_BATCH / 4, 256, 0, stream>>>(emb_w, lS_i, Tb);

    const dim3 blk(128);
    // Bottom MLP (all ReLU). Last layer writes row 0 of T (ldo = 1728).
    gemm_relu_wmma<<<dim3(B_BATCH / 64, 8), blk, 0, stream>>>(Xpad, 32, W0pad, bb0, A1, 512, 32, 512);
    gemm_relu_wmma<<<dim3(B_BATCH / 64, 4), blk, 0, stream>>>(A1, 512, Wb1, bb1, A2, 256, 512, 256);
    gemm_relu_wmma<<<dim3(B_BATCH / 64, 1), blk, 0, stream>>>(A2, 256, Wb2, bb2, Tb, T_STRIDE, 256, 64);

    // Pairwise interactions -> R [4096 x 416].
    interact_kernel<<<B_BATCH, 128, 0, stream>>>(Tb, R);

    // Top MLP: 416(pad)->512->256 with ReLU, then 256->1 + sigmoid.
    gemm_relu_wmma<<<dim3(B_BATCH / 64, 8), blk, 0, stream>>>(R, R_DIM, Wt0pad, bt0, A1, 512, R_DIM, 512);
    gemm_relu_wmma<<<dim3(B_BATCH / 64, 4), blk, 0, stream>>>(A1, 512, Wt1, bt1, A2, 256, 512, 256);
    top_final<<<B_BATCH / 8, 256, 0, stream>>>(A2, Wt2, bt2, out);
}